// New1_77395310674430
// MI455X (gfx1250) — compile-verified
//
#include <hip/hip_runtime.h>

// Fused geo-attention scorer for MI455X (gfx1250, wave32).
//
// Strategy (see analysis): one workgroup (256 thr = 8 waves) per batch row.
//  1. Gather hist[b] (200x128 fp32, concat of two 64-wide embedding rows) into LDS.
//  2. q = Wq @ targ and u = Wv^T @ targ  (u eliminates the entire v-GEMM:
//     v[b,s]·targ == hist[b,s]·u).
//  3. k = hist @ Wk^T via V_WMMA_F32_16X16X4_F32: wave w owns N-tile w (Wk
//     fragments hoisted to registers), loops 13 M-tiles x 32 chained WMMAs.
//     Each D element is scattered straight into scores[h] (raw-reshape law:
//     flat index i = s*128+f contributes q[i/200] * val to scores[i%200])
//     with ds_add_f32 — k is never materialized.
//  4. mask, exp, sum^0.5, then pred[b] = sum_h attn[h]*(hist_h·u) + hvr[h]*(hist_h·targ).

typedef __attribute__((ext_vector_type(2))) float v2f;
typedef __attribute__((ext_vector_type(8))) float v8f;

#define B_SZ   2048
#define H_LEN  200
#define E_DIM  128
#define MTILES 13            // ceil(200/16)
#define SM_HIST (208 * 128)  // hist padded to 13 M-tiles

__global__ __launch_bounds__(256)
void fused_geo_attn(const int* __restrict__ history,
                    const int* __restrict__ target,
                    const int* __restrict__ history_region,
                    const int* __restrict__ target_region,
                    const float* __restrict__ hvr,
                    const float* __restrict__ embT,
                    const float* __restrict__ embR,
                    const float* __restrict__ Wq,
                    const float* __restrict__ Wk,
                    const float* __restrict__ Wv,
                    float* __restrict__ out)
{
    extern __shared__ float smem[];
    float* s_hist = smem;                 // 208*128
    float* s_sc   = smem + SM_HIST;       // 208   scores -> exp_A
    float* s_q    = s_sc + 208;           // 128
    float* s_u    = s_q + 128;            // 128
    float* s_targ = s_u + 128;            // 128
    float* s_red  = s_targ + 128;         // 16: [0..7] wave partials, [8] pred, [9] 1/sqrt(sum)

    const int tid  = threadIdx.x;
    const int b    = blockIdx.x;
    const int wave = tid >> 5;
    const int lane = tid & 31;
    const int lo   = lane & 15;
    const int hi   = lane >> 4;

    const int  tgt = target[b];
    const int  tr  = target_region[b];
    const int* uh  = history + b * H_LEN;
    const int* hr  = history_region + b * H_LEN;

    // ---- stage 1: init + gathers ------------------------------------------
    if (tid < 208) s_sc[tid] = 0.f;
    if (tid < 16)  s_red[tid] = 0.f;
    if (tid < 32) {
        float4 vv = (tid < 16)
            ? ((const float4*)(embT + (size_t)tgt * 64))[tid]
            : ((const float4*)(embR + (size_t)tr  * 64))[tid - 16];
        ((float4*)s_targ)[tid] = vv;
    }
    for (int t = tid; t < H_LEN * 32; t += 256) {  // 32 float4 chunks per row
        int h = t >> 5, c = t & 31;
        float4 vv;
        if (c < 16) vv = ((const float4*)(embT + (size_t)uh[h] * 64))[c];
        else        vv = ((const float4*)(embR + (size_t)hr[h] * 64))[c - 16];
        ((float4*)(s_hist + h * E_DIM))[c] = vv;
    }
    __syncthreads();

    // ---- stage 2: q = Wq @ targ ; u = Wv^T @ targ -------------------------
    if (tid < 128) {
        const float* wr = Wq + tid * E_DIM;
        float acc = 0.f;
        #pragma unroll 8
        for (int e = 0; e < E_DIM; ++e) acc += wr[e] * s_targ[e];
        s_q[tid] = acc;
    } else {
        const int e0 = tid - 128;
        float acc = 0.f;
        #pragma unroll 8
        for (int f = 0; f < E_DIM; ++f) acc += Wv[f * E_DIM + e0] * s_targ[f];
        s_u[e0] = acc;
    }
    __syncthreads();

    // ---- stage 3: k-GEMM via WMMA + reshape-scatter into scores -----------
    {
        const int n0 = wave * 16;                        // wave owns one N-tile
        const float* wkrow = Wk + (size_t)(n0 + lo) * E_DIM;
        v2f bfr[32];                                     // hoisted Wk fragments
        #pragma unroll
        for (int s = 0; s < 32; ++s) {
            const int kk = 4 * s + 2 * hi;               // B[k][n] = Wk[n][k]
            bfr[s][0] = wkrow[kk];
            bfr[s][1] = wkrow[kk + 1];
        }
        for (int mt = 0; mt < MTILES; ++mt) {
            const int m0 = mt * 16;
            const float* arow = s_hist + (size_t)(m0 + lo) * E_DIM;
            v8f c = {};
            #pragma unroll
            for (int s = 0; s < 32; ++s) {
                const int kk = 4 * s + 2 * hi;
                v2f a;
                a[0] = arow[kk];
                a[1] = arow[kk + 1];
                c = __builtin_amdgcn_wmma_f32_16x16x4_f32(
                        false, a, false, bfr[s], (short)0, c, false, false);
            }
            #pragma unroll
            for (int r = 0; r < 8; ++r) {
                const int srow = m0 + r + 8 * hi;        // D row (history slot)
                if (srow < H_LEN) {
                    const int i = srow * E_DIM + n0 + lo; // flat kflat index
                    const int e = i / H_LEN;              // reshape: [128,200]
                    const int h = i - e * H_LEN;
                    unsafeAtomicAdd(&s_sc[h], s_q[e] * c[r]);   // ds_add_f32
                }
            }
        }
    }
    __syncthreads();

    // ---- stage 4: mask + exp + sum^0.5 ------------------------------------
    float val = 0.f;
    if (tid < H_LEN) {
        const float raw = s_sc[tid] * 0.08838834764831845f;  // 1/sqrt(128)
        val = (uh[tid] != tgt) ? __expf(raw) : 0.f;
        s_sc[tid] = val;                                     // exp_A
    }
    float p = val;
    #pragma unroll
    for (int off = 16; off > 0; off >>= 1) p += __shfl_xor(p, off);
    if (lane == 0) s_red[wave] = p;
    __syncthreads();
    if (tid == 0) {
        float sum = 0.f;
        #pragma unroll
        for (int w = 0; w < 8; ++w) sum += s_red[w];
        s_red[9] = 1.f / sqrtf(sum);                         // 1 / sum^BETA
    }
    __syncthreads();

    // ---- stage 5: pred = sum_h attn[h]*(hist_h . u) + hvr[h]*(hist_h . targ)
    const float inv = s_red[9];
    float acc = 0.f;
    for (int h = wave; h < H_LEN; h += 8) {
        const float* row = s_hist + h * E_DIM;
        float p1 = 0.f, p2 = 0.f;
        #pragma unroll
        for (int c4 = 0; c4 < 4; ++c4) {
            const int idx = lane + 32 * c4;
            const float x = row[idx];
            p1 += x * s_u[idx];
            p2 += x * s_targ[idx];
        }
        #pragma unroll
        for (int off = 16; off > 0; off >>= 1) {
            p1 += __shfl_xor(p1, off);
            p2 += __shfl_xor(p2, off);
        }
        if (lane == 0) acc += (s_sc[h] * inv) * p1 + hvr[h] * p2;
    }
    if (lane == 0) unsafeAtomicAdd(&s_red[8], acc);
    __syncthreads();
    if (tid == 0) out[b] = 1.f / (1.f + __expf(-s_red[8]));
}

extern "C" void kernel_launch(void* const* d_in, const int* in_sizes, int n_in,
                              void* d_out, int out_size, void* d_ws, size_t ws_size,
                              hipStream_t stream) {
    const int*   history = (const int*)d_in[0];
    const int*   target  = (const int*)d_in[1];
    const int*   hregion = (const int*)d_in[2];
    const int*   tregion = (const int*)d_in[3];
    const float* hvr     = (const float*)d_in[4];
    const float* embT    = (const float*)d_in[5];
    const float* embR    = (const float*)d_in[6];
    const float* Wq      = (const float*)d_in[7];
    const float* Wk      = (const float*)d_in[8];
    const float* Wv      = (const float*)d_in[9];
    float* out = (float*)d_out;

    const size_t lds = (size_t)(SM_HIST + 208 + 3 * 128 + 16) * sizeof(float); // ~109 KB
    fused_geo_attn<<<dim3(B_SZ), dim3(256), lds, stream>>>(
        history, target, hregion, tregion, hvr, embT, embR, Wq, Wk, Wv, out);
}